// BoxFilterND_6193342841707
// MI455X (gfx1250) — compile-verified
//
#include <hip/hip_runtime.h>
#include <cstdint>
#include <cstddef>

#ifndef __has_builtin
#define __has_builtin(x) 0
#endif

#define __AS1 __attribute__((address_space(1)))
#define __AS3 __attribute__((address_space(3)))

typedef __attribute__((ext_vector_type(2))) float v2f;
typedef __attribute__((ext_vector_type(8))) float v8f;
typedef __attribute__((ext_vector_type(4))) int   v4i;

// ---------------------------------------------------------------------------
// CDNA5 async global->LDS staging (ASYNCcnt path, CDNA5 ISA)
// ---------------------------------------------------------------------------
__device__ __forceinline__ void async_copy_b128(const float* gp, float* lp) {
#if __has_builtin(__builtin_amdgcn_global_load_async_to_lds_b128)
    __builtin_amdgcn_global_load_async_to_lds_b128(
        (__AS1 v4i*)gp, (__AS3 v4i*)lp, /*offset=*/0, /*cpol=*/0);
#else
    const uint32_t loff = (uint32_t)(uintptr_t)(__AS3 float*)lp;
    asm volatile("global_load_async_to_lds_b128 %0, %1, off"
                 :: "v"(loff), "v"(gp) : "memory");
#endif
}

__device__ __forceinline__ void wait_async0() {
#if __has_builtin(__builtin_amdgcn_s_wait_asynccnt)
    __builtin_amdgcn_s_wait_asynccnt(0);
#else
    asm volatile("s_wait_asynccnt 0" ::: "memory");
#endif
}

#if __has_builtin(__builtin_amdgcn_wmma_f32_16x16x4_f32)
#define HAVE_WMMA_F32X4 1
#else
#define HAVE_WMMA_F32X4 0
#endif

// ---------------------------------------------------------------------------
// Fused separable 17-tap box-sum, zero-padded borders.
// Grid: (64 tiles, 256 images). Block: 256 threads (8 wave32).
// ---------------------------------------------------------------------------
__global__ __launch_bounds__(256) void box17_fused(const float* __restrict__ in,
                                                   float* __restrict__ out) {
    constexpr int W       = 512;
    constexpr int TILE    = 64;
    constexpr int HALO    = 8;
    constexpr int IN_DIM  = TILE + 2 * HALO;  // 80
    constexpr int IN_P    = 80;               // float pitch of staged input (320B, 16B-aligned)
    constexpr int H_P     = 65;               // hsum pitch (+1 to kill bank conflicts)

    __shared__ float s_in[IN_DIM * IN_P];     // 25.6 KB
    __shared__ float s_h [IN_DIM * H_P];      // 20.8 KB

    const int t   = threadIdx.x;
    const int img = blockIdx.y;                      // b*c flattened, 0..255
    const int ty0 = (int)(blockIdx.x >> 3) << 6;     // tile row origin
    const int tx0 = (int)(blockIdx.x & 7) << 6;      // tile col origin
    const float* src = in  + (size_t)img * (W * W);
    float*       dst = out + (size_t)img * (W * W);

    // Zero-pad halo only for border tiles (uniform branch).
    const bool edge = (tx0 == 0) | (ty0 == 0) | (tx0 + TILE == W) | (ty0 + TILE == W);
    if (edge) {
        for (int i = t; i < IN_DIM * IN_P; i += 256) s_in[i] = 0.0f;
        __syncthreads();   // DS stores visible before async engine writes same LDS
    }

    // ---- Stage 80x80 input tile: 80 rows x 20 chunks of 16B, async to LDS ----
    for (int i = t; i < IN_DIM * 20; i += 256) {
        const int r  = i / 20;
        const int c4 = (i % 20) << 2;
        const int gr = ty0 - HALO + r;
        const int gc = tx0 - HALO + c4;          // chunks are wholly in or wholly out
        if ((unsigned)gr < (unsigned)W && (unsigned)gc < (unsigned)W) {
            async_copy_b128(src + (size_t)gr * W + gc, &s_in[r * IN_P + c4]);
        }
    }
    wait_async0();
    __syncthreads();

    // ---- Horizontal pass (dim 3): running 17-tap sum per row ----
    // 160 active threads: (row 0..79) x (half 0..1), 32 outputs each.
    if (t < 160) {
        const int r  = t >> 1;
        const int c0 = (t & 1) << 5;
        const float* row  = &s_in[r * IN_P + c0];  // out col c <- in cols c..c+16
        float*       hrow = &s_h [r * H_P  + c0];
        float s = 0.0f;
        #pragma unroll
        for (int j = 0; j < 17; ++j) s += row[j];
        #pragma unroll
        for (int i = 0; i < 32; ++i) {
            hrow[i] = s;
            if (i < 31) s += row[i + 17] - row[i];
        }
    }
    __syncthreads();

    // ---- Vertical pass (dim 2) ----
#if HAVE_WMMA_F32X4
    // Banded matmul on the matrix pipe: Out(16x16) = sum_k Band(16x4) * Hsum(4x16).
    // Band[m][k] = 1.0 iff k in [m, m+16] (17-row window), weights exact in f32.
    const int lane = t & 31;
    const int wave = t >> 5;
    const int hi   = lane >> 4;    // lane half: ISA A/B/D layouts split at lane 16
    const int m    = lane & 15;
    #pragma unroll
    for (int ti = 0; ti < 2; ++ti) {
        const int tile = wave * 2 + ti;          // 16 tiles of 16x16, 2 per wave
        const int mt = tile >> 2, nt = tile & 3;
        const int col = (nt << 4) + m;           // N = lane%16
        v8f acc = {};
        #pragma unroll
        for (int kb = 0; kb < 32; kb += 4) {     // K = 32 hsum rows per output row-tile
            const int k0 = kb + (hi << 1);       // A/B: vgpr0 -> K{0,2}, vgpr1 -> K{1,3}
            v2f a, b;
            a.x = ((unsigned)(k0     - m) <= 16u) ? 1.0f : 0.0f;
            a.y = ((unsigned)(k0 + 1 - m) <= 16u) ? 1.0f : 0.0f;
            const int r0 = (mt << 4) + k0;
            b.x = s_h[ r0      * H_P + col];
            b.y = s_h[(r0 + 1) * H_P + col];
            acc = __builtin_amdgcn_wmma_f32_16x16x4_f32(
                      false, a, false, b, (short)0, acc, false, false);
        }
        // D layout: VGPR v -> M = v + 8*hi, N = lane%16
        float* dp = dst + (size_t)(ty0 + (mt << 4) + (hi << 3)) * W + (tx0 + col);
        #pragma unroll
        for (int v = 0; v < 8; ++v) dp[(size_t)v * W] = acc[v];
    }
#else
    // VALU fallback: running 17-tap column sum. 256 threads: (col 0..63) x (seg 0..3).
    const int x   = t & 63;
    const int r0  = (t >> 6) << 4;
    float s = 0.0f;
    #pragma unroll
    for (int j = 0; j < 17; ++j) s += s_h[(r0 + j) * H_P + x];
    #pragma unroll
    for (int i = 0; i < 16; ++i) {
        dst[(size_t)(ty0 + r0 + i) * W + tx0 + x] = s;
        if (i < 15) s += s_h[(r0 + i + 17) * H_P + x] - s_h[(r0 + i) * H_P + x];
    }
#endif
}

// ---------------------------------------------------------------------------
extern "C" void kernel_launch(void* const* d_in, const int* in_sizes, int n_in,
                              void* d_out, int out_size, void* d_ws, size_t ws_size,
                              hipStream_t stream) {
    (void)in_sizes; (void)n_in; (void)out_size; (void)d_ws; (void)ws_size;
    const float* in = (const float*)d_in[0];
    float* out      = (float*)d_out;
    // (8,32,512,512): 256 images, 8x8 tiles of 64x64 each.
    dim3 grid(64, 256, 1);
    dim3 block(256, 1, 1);
    box17_fused<<<grid, block, 0, stream>>>(in, out);
}